// DiffPool_56573309223698
// MI455X (gfx1250) — compile-verified
//
#include <hip/hip_runtime.h>
#include <hip/hip_bf16.h>
#include <math.h>

// ---------------- problem dims (fixed by the reference) ----------------
#define NN   16384   // nodes
#define CIN  128     // input channels
#define KCL  512     // clusters (OUT_CH)

typedef float v2f __attribute__((ext_vector_type(2)));
typedef float v8f __attribute__((ext_vector_type(8)));
typedef int   gvec4i __attribute__((vector_size(16)));  // matches builtin's b128 type

#define GLOBAL_AS __attribute__((address_space(1)))
#define LDS_AS    __attribute__((address_space(3)))

// f32 WMMA: D(16x16 f32) = A(16x4 f32) * B(4x16 f32) + C
__device__ __forceinline__ v8f wmma_f32x4(v2f a, v2f b, v8f c) {
    return __builtin_amdgcn_wmma_f32_16x16x4_f32(
        /*neg_a=*/false, a, /*neg_b=*/false, b,
        /*c_mod=*/(short)0, c, /*reuse_a=*/false, /*reuse_b=*/false);
}

// async memory->LDS copy (gfx1250 GLOBAL_LOAD_ASYNC_TO_LDS_*), ASYNCcnt-tracked
__device__ __forceinline__ void async_ld_b32(const float* g, void* l) {
    __builtin_amdgcn_global_load_async_to_lds_b32(
        (GLOBAL_AS int*)g, (LDS_AS int*)l, 0, 0);
}
__device__ __forceinline__ void async_ld_b128(const float* g, void* l) {
    __builtin_amdgcn_global_load_async_to_lds_b128(
        (GLOBAL_AS gvec4i*)g, (LDS_AS gvec4i*)l, 0, 0);
}
__device__ __forceinline__ void wait_async0() {
    asm volatile("s_wait_asynccnt 0x0" ::: "memory");
}

// ---------------- kernel 0: zero workspace (T and M, contiguous) -------
__global__ __launch_bounds__(256) void k_zero(float* __restrict__ p, size_t n) {
    size_t i = (size_t)blockIdx.x * 256u + threadIdx.x;
    if (i < n) p[i] = 0.0f;
}

// ---------------- kernel 1: S = softmax(x@Wa+ba), h = x@Wf+bf ----------
// one block (8 waves) per 16-node tile
__global__ __launch_bounds__(256) void k_s_h(const float* __restrict__ x,
                                             const float* __restrict__ Wa,
                                             const float* __restrict__ ba,
                                             const float* __restrict__ Wf,
                                             const float* __restrict__ bf,
                                             float* __restrict__ S,
                                             float* __restrict__ h) {
    // CIN+4 padding: row stride 528B (16B aligned for async B128), 4-bank skew
    __shared__ float xs[16][CIN + 4];
    __shared__ float lg[16][KCL + 1];

    const int n0    = blockIdx.x * 16;
    const int tid   = threadIdx.x;
    const int lane  = tid & 31;
    const int wave  = tid >> 5;
    const int m     = lane & 15;
    const int khalf = lane >> 4;

    // stage x tile [16 x 128] via async B128 (512 float4 chunks / 256 threads)
    for (int i = tid; i < 16 * (CIN / 4); i += 256) {
        const int row = i >> 5, c4 = i & 31;
        async_ld_b128(&x[(size_t)(n0 + row) * CIN + c4 * 4], &xs[row][c4 * 4]);
    }
    wait_async0();
    __syncthreads();

    // logits: each wave owns 4 cluster tiles (64 columns)
    for (int t = 0; t < 4; ++t) {
        const int c0 = wave * 64 + t * 16;
        v8f acc = {};
        for (int kk = 0; kk < CIN; kk += 4) {
            v2f a, b;
            a.x = xs[m][kk + 2 * khalf];
            a.y = xs[m][kk + 2 * khalf + 1];
            b.x = Wa[(size_t)(kk + 2 * khalf)     * KCL + c0 + m];
            b.y = Wa[(size_t)(kk + 2 * khalf + 1) * KCL + c0 + m];
            acc = wmma_f32x4(a, b, acc);
        }
        const float bias = ba[c0 + m];
#pragma unroll
        for (int v = 0; v < 8; ++v)
            lg[v + 8 * khalf][c0 + m] = acc[v] + bias;
    }
    __syncthreads();

    // wave-level softmax: wave handles rows 2w, 2w+1 (512 wide)
    for (int r = wave * 2; r < wave * 2 + 2; ++r) {
        float mx = -INFINITY;
        for (int j = lane; j < KCL; j += 32) mx = fmaxf(mx, lg[r][j]);
#pragma unroll
        for (int s = 16; s; s >>= 1) mx = fmaxf(mx, __shfl_xor(mx, s, 32));
        float sum = 0.0f;
        for (int j = lane; j < KCL; j += 32) {
            float e = __expf(lg[r][j] - mx);
            lg[r][j] = e;
            sum += e;
        }
#pragma unroll
        for (int s = 16; s; s >>= 1) sum += __shfl_xor(sum, s, 32);
        const float inv = 1.0f / sum;
        for (int j = lane; j < KCL; j += 32)
            S[(size_t)(n0 + r) * KCL + j] = lg[r][j] * inv;
    }

    // h: each wave owns one 16-channel tile
    {
        const int c0 = wave * 16;
        v8f acc = {};
        for (int kk = 0; kk < CIN; kk += 4) {
            v2f a, b;
            a.x = xs[m][kk + 2 * khalf];
            a.y = xs[m][kk + 2 * khalf + 1];
            b.x = Wf[(size_t)(kk + 2 * khalf)     * CIN + c0 + m];
            b.y = Wf[(size_t)(kk + 2 * khalf + 1) * CIN + c0 + m];
            acc = wmma_f32x4(a, b, acc);
        }
        const float bias = bf[c0 + m];
#pragma unroll
        for (int v = 0; v < 8; ++v)
            h[(size_t)(n0 + v + 8 * khalf) * CIN + c0 + m] = acc[v] + bias;
    }
}

// ---------------- kernel 2: T[row,:] += w * S[col,:] (edge scatter) ----
// one wave per edge; S row read as 4x global_load_b128 per lane.
// S and T live in the 192MB L2, so atomics are L2-resident.
__global__ __launch_bounds__(256) void k_scatter(const long long* __restrict__ ei,
                                                 const float* __restrict__ ew,
                                                 const float* __restrict__ S,
                                                 float* __restrict__ T,
                                                 int nEdges) {
    const int e = blockIdx.x * 8 + (threadIdx.x >> 5);
    if (e >= nEdges) return;
    const int lane = threadIdx.x & 31;
    const long long r = ei[e];           // edge_index[0][e]
    const long long c = ei[nEdges + e];  // edge_index[1][e]
    const float w = ew[e];
    const float4* __restrict__ srow4 = (const float4*)(S + (size_t)c * KCL);
    float* __restrict__ trow         = T + (size_t)r * KCL;
    __builtin_prefetch(srow4 + lane, 0, 0);   // global_prefetch_b8
#pragma unroll
    for (int jj = 0; jj < 4; ++jj) {
        const float4 sv = srow4[jj * 32 + lane];
        const int base  = (jj * 32 + lane) * 4;
        atomicAdd(&trow[base + 0], w * sv.x);
        atomicAdd(&trow[base + 1], w * sv.y);
        atomicAdd(&trow[base + 2], w * sv.z);
        atomicAdd(&trow[base + 3], w * sv.w);
    }
}

// ---------------- kernel 3: pooled_x = S^T @ h  [512 x 128] ------------
// one block per 16-cluster strip; wave w owns channel tile 16w.
// S chunk transposed in-flight by per-lane async-to-LDS destinations.
__global__ __launch_bounds__(256) void k_pooled_x(const float* __restrict__ S,
                                                  const float* __restrict__ h,
                                                  float* __restrict__ px) {
    __shared__ float sT[16][17];  // padded
    const int k0    = blockIdx.x * 16;
    const int tid   = threadIdx.x;
    const int lane  = tid & 31;
    const int wave  = tid >> 5;
    const int m     = lane & 15;
    const int khalf = lane >> 4;
    const int c0    = wave * 16;
    const int si = tid >> 4, sj = tid & 15;

    v8f acc = {};
    for (int n0 = 0; n0 < NN; n0 += 16) {
        __syncthreads();
        async_ld_b32(&S[(size_t)(n0 + si) * KCL + k0 + sj], &sT[sj][si]);
        wait_async0();
        __syncthreads();
#pragma unroll
        for (int kk = 0; kk < 16; kk += 4) {
            v2f a, b;
            a.x = sT[m][kk + 2 * khalf];
            a.y = sT[m][kk + 2 * khalf + 1];
            b.x = h[(size_t)(n0 + kk + 2 * khalf)     * CIN + c0 + m];
            b.y = h[(size_t)(n0 + kk + 2 * khalf + 1) * CIN + c0 + m];
            acc = wmma_f32x4(a, b, acc);
        }
    }
#pragma unroll
    for (int v = 0; v < 8; ++v)
        px[(size_t)(k0 + v + 8 * khalf) * CIN + c0 + m] = acc[v];
}

// ---------------- kernel 4: M = S^T @ T  [512 x 512], 4-way N split ----
__global__ __launch_bounds__(256) void k_M(const float* __restrict__ S,
                                           const float* __restrict__ T,
                                           float* __restrict__ M) {
    __shared__ float sT[16][17];
    const int k0    = blockIdx.x * 16;
    const int nbase = blockIdx.y * (NN / 4);
    const int tid   = threadIdx.x;
    const int lane  = tid & 31;
    const int wave  = tid >> 5;
    const int m     = lane & 15;
    const int khalf = lane >> 4;
    const int si = tid >> 4, sj = tid & 15;

    v8f acc[4] = {};
    for (int n0 = nbase; n0 < nbase + NN / 4; n0 += 16) {
        __syncthreads();
        async_ld_b32(&S[(size_t)(n0 + si) * KCL + k0 + sj], &sT[sj][si]);
        wait_async0();
        __syncthreads();
#pragma unroll
        for (int kk = 0; kk < 16; kk += 4) {
            v2f a;
            a.x = sT[m][kk + 2 * khalf];
            a.y = sT[m][kk + 2 * khalf + 1];
#pragma unroll
            for (int t = 0; t < 4; ++t) {  // wave owns 64 output columns
                const int c0 = wave * 64 + t * 16;
                v2f b;
                b.x = T[(size_t)(n0 + kk + 2 * khalf)     * KCL + c0 + m];
                b.y = T[(size_t)(n0 + kk + 2 * khalf + 1) * KCL + c0 + m];
                acc[t] = wmma_f32x4(a, b, acc[t]);
            }
        }
    }
#pragma unroll
    for (int t = 0; t < 4; ++t) {
        const int c0 = wave * 64 + t * 16;
#pragma unroll
        for (int v = 0; v < 8; ++v)
            atomicAdd(&M[(size_t)(k0 + v + 8 * khalf) * KCL + c0 + m], acc[t][v]);
    }
}

// ---------------- kernel 5: pooled_adj = M + M^T -----------------------
__global__ __launch_bounds__(256) void k_adj(const float* __restrict__ M,
                                             float* __restrict__ pa) {
    const int i = blockIdx.x * 256 + threadIdx.x;
    if (i < KCL * KCL) {
        const int r = i >> 9, c = i & (KCL - 1);
        pa[i] = M[i] + M[(size_t)c * KCL + r];
    }
}

// ---------------- launcher --------------------------------------------
extern "C" void kernel_launch(void* const* d_in, const int* in_sizes, int n_in,
                              void* d_out, int out_size, void* d_ws, size_t ws_size,
                              hipStream_t stream) {
    (void)n_in; (void)out_size; (void)ws_size;
    const float*     x  = (const float*)d_in[0];
    const long long* ei = (const long long*)d_in[1];  // int64 edge_index [2, E]
    const float*     ew = (const float*)d_in[2];
    const float*     Wa = (const float*)d_in[3];
    const float*     ba = (const float*)d_in[4];
    const float*     Wf = (const float*)d_in[5];
    const float*     bf = (const float*)d_in[6];

    float* out  = (float*)d_out;
    float* px   = out;                              // [512,128]
    float* padj = out + (size_t)KCL * CIN;          // [512,512]
    float* S    = padj + (size_t)KCL * KCL;         // [16384,512]

    float* h = (float*)d_ws;                        // [16384,128]  8 MB
    float* T = h + (size_t)NN * CIN;                // [16384,512] 32 MB
    float* M = T + (size_t)NN * KCL;                // [512,512]    1 MB

    const int nEdges = in_sizes[2];                 // 524288

    const size_t zc = (size_t)NN * KCL + (size_t)KCL * KCL;
    k_zero<<<(unsigned)((zc + 255) / 256), 256, 0, stream>>>(T, zc);

    k_s_h<<<NN / 16, 256, 0, stream>>>(x, Wa, ba, Wf, bf, S, h);
    k_scatter<<<(nEdges + 7) / 8, 256, 0, stream>>>(ei, ew, S, T, nEdges);
    k_pooled_x<<<KCL / 16, 256, 0, stream>>>(S, h, px);
    dim3 gM(KCL / 16, 4);
    k_M<<<gM, 256, 0, stream>>>(S, T, M);
    k_adj<<<(KCL * KCL + 255) / 256, 256, 0, stream>>>(M, padj);
}